// Qwen3MoeSparseFusedMoeBlock_21638045237564
// MI455X (gfx1250) — compile-verified
//
#include <hip/hip_runtime.h>
#include <hip/hip_bf16.h>

// Problem sizes (compile-time, from the reference)
#define E_  64      // experts
#define H_  2048    // hidden
#define I_  768     // moe intermediate
#define K_  8       // top-k
#define T_  8192    // tokens = B*S = 4*2048

// MoE kernel tiling
#define CH   64     // H-chunk of the X tile staged in LDS (phase 1)
#define NCH  (H_ / CH)
#define SXS  (CH + 4)   // padded LDS row stride for s_x (68 % 64 == 4 -> conflict-free)
#define SHS  772        // padded LDS row stride for s_h (772 % 64 == 4 -> conflict-free)

typedef float v2f __attribute__((ext_vector_type(2)));
typedef float v8f __attribute__((ext_vector_type(8)));
typedef unsigned int v4u __attribute__((ext_vector_type(4)));
typedef int v4i __attribute__((ext_vector_type(4)));
typedef int v8i __attribute__((ext_vector_type(8)));

// Full-precision fp32 matrix op: D = A(16x4) * B(4x16) + C(16x16)
static __device__ __forceinline__ v8f wmma_f32_16x16x4(v2f a, v2f b, v8f c) {
  return __builtin_amdgcn_wmma_f32_16x16x4_f32(false, a, false, b, (short)0, c,
                                               false, false);
}

// ---------------------------------------------------------------------------
// TDM gather-mode load: pull 16 rows (16-bit indices, padded rows = 0xFFFF ->
// OOB -> zero-filled) of CH fp32 elements each from a row-major [T_, H_]
// tensor into LDS, inserting 4 pad DWORDs per row (stride CH+4, bank-safe).
// Descriptor per CDNA5 ISA 8.3-8.7; this toolchain's 6-arg builtin.
// ---------------------------------------------------------------------------
static __device__ __forceinline__ void tdm_gather16(unsigned lds_addr,
                                                    const float* gsrc,
                                                    const unsigned* pk) {
  const unsigned long long ga = (unsigned long long)(uintptr_t)gsrc;
  v4u g0;
  g0.x = 0x80000001u;                         // count=1, gather_mode=1, idx16
  g0.y = lds_addr;                            // LDS destination (bytes)
  g0.z = (unsigned)(ga & 0xFFFFFFFFu);        // global_addr[31:0]
  g0.w = (unsigned)((ga >> 32) & 0x01FFFFFFu) // global_addr[56:32]
         | 0x80000000u;                       // type=2 ("image")
  v8i g1;
  g1[0] = (int)((2u << 16)      // data_size = 4B
                | (1u << 20)    // pad_enable
                | (5u << 22)    // pad_interval: every 64 DWORDs (one row)
                | (3u << 25));  // pad_amount: 4 DWORDs
  g1[1] = (int)((unsigned)CH << 16);       // tensor_dim0 (low16)
  g1[2] = (int)((unsigned)T_ << 16);       // tensor_dim1 (low16) | dim0 hi = 0
  g1[3] = (int)((unsigned)CH << 16);       // tile_dim0 | dim1 hi = 0
  g1[4] = 16;                              // tile_dim1 = #row indices
  g1[5] = H_;                              // tensor_dim0_stride (elements)
  g1[6] = 0;
  g1[7] = 0;
  v4i g2, g3;
  g2[0] = (int)pk[0]; g2[1] = (int)pk[1]; g2[2] = (int)pk[2]; g2[3] = (int)pk[3];
  g3[0] = (int)pk[4]; g3[1] = (int)pk[5]; g3[2] = (int)pk[6]; g3[3] = (int)pk[7];
  v8i g4;
  g4[0] = 0; g4[1] = 0; g4[2] = 0; g4[3] = 0;
  g4[4] = 0; g4[5] = 0; g4[6] = 0; g4[7] = 0;   // unused trailing group
  __builtin_amdgcn_tensor_load_to_lds(g0, g1, g2, g3, g4, 0);
}

// ---------------------------------------------------------------------------
// Kernel 0: zero the output accumulation region + expert counts
// ---------------------------------------------------------------------------
__global__ void zero_kernel(float* __restrict__ out, int* __restrict__ counts) {
  size_t i = (size_t)blockIdx.x * blockDim.x + threadIdx.x;
  const size_t stride = (size_t)gridDim.x * blockDim.x;
  const size_t n = (size_t)T_ * H_;
  for (; i < n; i += stride) out[i] = 0.f;
  if (blockIdx.x == 0 && threadIdx.x < E_) counts[threadIdx.x] = 0;
}

// ---------------------------------------------------------------------------
// Kernel 1: router logits + softmax + top-8 + renormalize.  One wave / token.
// ---------------------------------------------------------------------------
__global__ void router_topk_kernel(const float* __restrict__ x,
                                   const float* __restrict__ gw,
                                   float* __restrict__ logits_out,
                                   int* __restrict__ counts,
                                   int* __restrict__ tok_sel,
                                   float* __restrict__ tok_w) {
  const int wave = threadIdx.x >> 5;
  const int lane = threadIdx.x & 31;
  const int t = blockIdx.x * 4 + wave;

  __shared__ float s_lg[4][E_];
  __shared__ int   s_sel[4][K_];
  __shared__ float s_sv[4][K_];

  float xr[H_ / 32];
  const float* xp = x + (size_t)t * H_ + lane;
#pragma unroll
  for (int i = 0; i < H_ / 32; ++i) xr[i] = xp[i * 32];

  for (int e = 0; e < E_; ++e) {
    const float* g = gw + (size_t)e * H_ + lane;
    float acc = 0.f;
#pragma unroll
    for (int i = 0; i < H_ / 32; ++i) acc = fmaf(xr[i], g[i * 32], acc);
#pragma unroll
    for (int off = 16; off > 0; off >>= 1) acc += __shfl_xor(acc, off, 32);
    if (lane == 0) s_lg[wave][e] = acc;
  }

  float l0 = s_lg[wave][lane];
  float l1 = s_lg[wave][lane + 32];
  logits_out[(size_t)t * E_ + lane] = l0;       // raw logits are an output
  logits_out[(size_t)t * E_ + lane + 32] = l1;

  float mx = fmaxf(l0, l1);
#pragma unroll
  for (int off = 16; off > 0; off >>= 1) mx = fmaxf(mx, __shfl_xor(mx, off, 32));
  float p0 = __expf(l0 - mx);
  float p1 = __expf(l1 - mx);

#pragma unroll
  for (int k = 0; k < K_; ++k) {
    float v = fmaxf(p0, p1);
    int vi = (p0 >= p1) ? lane : lane + 32;
#pragma unroll
    for (int off = 16; off > 0; off >>= 1) {
      float ov = __shfl_xor(v, off, 32);
      int   oi = __shfl_xor(vi, off, 32);
      if (ov > v || (ov == v && oi < vi)) { v = ov; vi = oi; }
    }
    if (lane == 0) { s_sel[wave][k] = vi; s_sv[wave][k] = v; }
    if (vi == lane)           p0 = -1.f;
    else if (vi == lane + 32) p1 = -1.f;
  }

  if (lane == 0) {
    float ws = 0.f;
    for (int k = 0; k < K_; ++k) ws += s_sv[wave][k];
    const float inv = 1.f / ws;
    for (int k = 0; k < K_; ++k) {
      const int e = s_sel[wave][k];
      const int idx = t * K_ + k;
      tok_sel[idx] = e;
      tok_w[idx] = s_sv[wave][k] * inv;
      atomicAdd(&counts[e], 1);
    }
  }
}

// ---------------------------------------------------------------------------
// Kernel 2: exclusive prefix sum over 64 expert counts
// ---------------------------------------------------------------------------
__global__ void scan_kernel(const int* __restrict__ counts,
                            int* __restrict__ offsets,
                            int* __restrict__ cursor) {
  if (threadIdx.x == 0) {
    int run = 0;
    for (int e = 0; e < E_; ++e) {
      offsets[e] = run;
      cursor[e] = run;
      run += counts[e];
    }
    offsets[E_] = run;
  }
}

// ---------------------------------------------------------------------------
// Kernel 3: scatter (token, weight) into per-expert segments
// ---------------------------------------------------------------------------
__global__ void scatter_kernel(const int* __restrict__ tok_sel,
                               const float* __restrict__ tok_w,
                               int* __restrict__ cursor,
                               int* __restrict__ exp_tok,
                               float* __restrict__ exp_w) {
  const int idx = blockIdx.x * blockDim.x + threadIdx.x;
  if (idx >= T_ * K_) return;
  const int e = tok_sel[idx];
  const int pos = atomicAdd(&cursor[e], 1);
  exp_tok[pos] = idx >> 3;
  exp_w[pos] = tok_w[idx];
}

// ---------------------------------------------------------------------------
// Kernel 4: fused expert MLP on 16-token tiles.  X tile staged by the Tensor
// Data Mover (gather mode, double-buffered, overlapped with WMMA compute);
// all matmuls via v_wmma_f32_16x16x4_f32.  Weight loads indexed with integer
// offsets off the __restrict__ kernel args so they stay GLOBAL_LOAD (not
// flat) and the tile/k offsets fold into the 24-bit instruction offset.
// ---------------------------------------------------------------------------
__global__ void __launch_bounds__(256)
moe_expert_kernel(const float* __restrict__ x,
                  const float* __restrict__ gate_w,   // [E, I, H]
                  const float* __restrict__ up_w,     // [E, I, H]
                  const float* __restrict__ down_w,   // [E, H, I]
                  const int* __restrict__ counts,
                  const int* __restrict__ offsets,
                  const int* __restrict__ exp_tok,
                  const float* __restrict__ exp_w,
                  float* __restrict__ out) {
  const int e = blockIdx.y;
  const int cnt = counts[e];
  const int m0 = blockIdx.x * 16;
  if (m0 >= cnt) return;           // block-uniform exit (EXEC all-1 for WMMA)

  const int tid = threadIdx.x;
  const int wave = tid >> 5;       // 8 waves
  const int lane = tid & 31;
  const int n = lane & 15;                 // tile column / A row (M)
  const int khalf = (lane >> 4) << 1;      // K-pair select per ISA A/B layout
  const int hi2 = lane >> 4;               // C/D: M = r + 8*hi2

  __shared__ float s_x[2][16 * SXS];       // 2 x 4.25 KB, TDM double buffer
  __shared__ float s_h[16 * SHS];          // 48.3 KB SwiGLU intermediate
  __shared__ int   s_tok[16];
  __shared__ float s_wt[16];

  if (tid < 16) {
    const int j = m0 + tid;
    if (j < cnt) {
      s_tok[tid] = exp_tok[offsets[e] + j];
      s_wt[tid] = exp_w[offsets[e] + j];
    } else {
      s_tok[tid] = -1;                     // -> 0xFFFF row index -> TDM OOB -> 0
      s_wt[tid] = 0.f;
    }
  }
  __syncthreads();

  // Packed 16-bit gather indices (uniform; pin to SGPRs for the descriptor)
  unsigned pk[8];
#pragma unroll
  for (int j = 0; j < 8; ++j)
    pk[j] = (unsigned)__builtin_amdgcn_readfirstlane(
        (int)(((unsigned)s_tok[2 * j] & 0xFFFFu) |
              (((unsigned)s_tok[2 * j + 1] & 0xFFFFu) << 16)));
  const unsigned lds0 = (unsigned)(uintptr_t)&s_x[0][0];
  const unsigned lds1 = (unsigned)(uintptr_t)&s_x[1][0];

  // Per-expert bases rooted in the kernel args (keeps addrspace(1) inference)
  const size_t gu_base = (size_t)e * I_ * H_;
  const int wrow = (wave * 6 * 16 + n) * H_ + khalf;  // wave's first tile row

  // Persistent fp32 accumulators: 6 gate + 6 up 16x16 tiles per wave
  v8f accg[6], accu[6];
#pragma unroll
  for (int q = 0; q < 6; ++q)
#pragma unroll
    for (int r = 0; r < 8; ++r) { accg[q][r] = 0.f; accu[q][r] = 0.f; }

  // Prefetch chunk 0 via TDM
  if (wave == 0) tdm_gather16(lds0, x, pk);

  // ---- Phase 1: h_pre = X[16,H] @ {G,U}^T, H streamed via TDM double buffer
  for (int c = 0; c < NCH; ++c) {
    if (wave == 0) __builtin_amdgcn_s_wait_tensorcnt(0);
    __syncthreads();                       // chunk c visible; buf c^1 free
    if (wave == 0 && c + 1 < NCH)
      tdm_gather16((c & 1) ? lds0 : lds1, x + (size_t)(c + 1) * CH, pk);

    const float* sx = &s_x[c & 1][n * SXS + khalf];
    const float* gch = gate_w + gu_base + wrow + c * CH;  // chunk base (global)
    const float* uch = up_w + gu_base + wrow + c * CH;
#pragma unroll
    for (int kk = 0; kk < CH; kk += 4) {
      const v2f a = *(const v2f*)(sx + kk);
#pragma unroll
      for (int q = 0; q < 6; ++q) {
        // q*16*H_ + kk folds into the 24-bit global_load immediate offset
        const v2f bg = *(const v2f*)(gch + q * 16 * H_ + kk);
        const v2f bu = *(const v2f*)(uch + q * 16 * H_ + kk);
        accg[q] = wmma_f32_16x16x4(a, bg, accg[q]);
        accu[q] = wmma_f32_16x16x4(a, bu, accu[q]);
      }
    }
  }

  // ---- SwiGLU in-register, stage h[16, I] to LDS
#pragma unroll
  for (int q = 0; q < 6; ++q) {
    const int i0 = (wave * 6 + q) * 16;
#pragma unroll
    for (int r = 0; r < 8; ++r) {
      const float g = accg[q][r];
      const float u = accu[q][r];
      const float hval = (g / (1.f + __expf(-g))) * u;   // silu(g) * u
      s_h[(r + 8 * hi2) * SHS + i0 + n] = hval;
    }
  }
  __syncthreads();

  // ---- Phase 2: out_tile = h[16,I] @ D^T, scale by routing weight, scatter
  const size_t d_base = (size_t)e * H_ * I_;
  const float* shp = &s_h[n * SHS + khalf];
#pragma unroll 1
  for (int p = 0; p < 16; ++p) {
    const int h0 = (wave * 16 + p) * 16;
    const float* dptr = down_w + d_base + (size_t)((h0 + n) * I_ + khalf);
    v8f acc;
#pragma unroll
    for (int r = 0; r < 8; ++r) acc[r] = 0.f;
#pragma unroll 32
    for (int kk = 0; kk < I_; kk += 4) {
      const v2f a = *(const v2f*)(shp + kk);
      const v2f b = *(const v2f*)(dptr + kk);
      acc = wmma_f32_16x16x4(a, b, acc);
    }
#pragma unroll
    for (int r = 0; r < 8; ++r) {
      const int m = r + 8 * hi2;
      const int t = s_tok[m];
      if (t >= 0)
        atomicAdd(&out[(size_t)t * H_ + h0 + n], acc[r] * s_wt[m]);
    }
  }
}

// ---------------------------------------------------------------------------
extern "C" void kernel_launch(void* const* d_in, const int* in_sizes, int n_in,
                              void* d_out, int out_size, void* d_ws,
                              size_t ws_size, hipStream_t stream) {
  const float* x  = (const float*)d_in[0];   // [T, H]
  const float* gw = (const float*)d_in[1];   // [E, H]
  const float* gp = (const float*)d_in[2];   // [E, I, H]
  const float* up = (const float*)d_in[3];   // [E, I, H]
  const float* dp = (const float*)d_in[4];   // [E, H, I]

  float* out = (float*)d_out;                   // [T, H]
  float* logits = out + (size_t)T_ * H_;        // [T, E]

  int* counts  = (int*)d_ws;                 // [E]
  int* offsets = counts + E_;                // [E+1]
  int* cursor  = offsets + (E_ + 1);         // [E]
  int* tok_sel = cursor + E_;                // [T*K]
  float* tok_w = (float*)(tok_sel + T_ * K_);// [T*K]
  int* exp_tok = (int*)(tok_w + T_ * K_);    // [T*K]
  float* exp_w = (float*)(exp_tok + T_ * K_);// [T*K]

  zero_kernel<<<2048, 256, 0, stream>>>(out, counts);
  router_topk_kernel<<<T_ / 4, 128, 0, stream>>>(x, gw, logits, counts,
                                                 tok_sel, tok_w);
  scan_kernel<<<1, 32, 0, stream>>>(counts, offsets, cursor);
  scatter_kernel<<<(T_ * K_) / 256, 256, 0, stream>>>(tok_sel, tok_w, cursor,
                                                      exp_tok, exp_w);
  moe_expert_kernel<<<dim3(512, E_), 256, 0, stream>>>(
      x, gp, up, dp, counts, offsets, exp_tok, exp_w, out);
}